// ItemItemModel_15590731285238
// MI455X (gfx1250) — compile-verified
//
#include <hip/hip_runtime.h>

typedef __attribute__((ext_vector_type(2))) float v2f;
typedef __attribute__((ext_vector_type(8))) float v8f;

#define N_NODES 100000
#define N_EDGES 1600000

// ---------------------------------------------------------------------------
// Pass 1 (once): expbuf[e] = exp(attr[e]);  s[dst[e]] += expbuf[e]
// (segment-max subtraction cancels exactly in softmax; attr in [0,1) so exp
// is well-conditioned — mathematically identical to the reference)
// ---------------------------------------------------------------------------
__global__ void edge_expsum(const float* __restrict__ attr,
                            const long long* __restrict__ ei, // [2, E]
                            float* __restrict__ expbuf,
                            float* __restrict__ s, int nE) {
    int e = blockIdx.x * blockDim.x + threadIdx.x;
    if (e >= nE) return;
    float v = __expf(attr[e]);
    expbuf[e] = v;
    int d = (int)ei[(size_t)nE + e];
    atomicAdd(&s[d], v);
}

// ---------------------------------------------------------------------------
// XW = X @ W^T.  X:[nrows,K], W:[64,K], XW:[nrows,64].  K = 128 or 64.
// Block: 256 threads = 8 waves -> 32 rows x 64 cols.  Each wave computes a
// 16x16 tile with K/4 chained V_WMMA_F32_16X16X4_F32 (full f32 precision).
// LDS rows padded to K+2 floats: bank-conflict-free and 8B aligned fragments.
//
// f32 16x4 A fragment (per ISA 7.12.2): lane l -> m = l&15, kpair = (l>>4)*2,
//   a = A[m][k+kpair .. k+kpair+1].  B (4x16) mirrors: b = B[k+kpair..][n=l&15],
//   i.e. with B = W^T:  b = W[n][k+kpair .. +1].
// C/D: lane l holds col n=l&15, rows (l<16 ? 0..7 : 8..15) in v[0..7].
// ---------------------------------------------------------------------------
template <int K>
__global__ void gemm_xwT(const float* __restrict__ X,
                         const float* __restrict__ W,
                         float* __restrict__ XW, int nrows) {
    constexpr int LDK = K + 2;
    __shared__ float sA[32 * LDK];
    __shared__ float sB[64 * LDK];

    const int tid = threadIdx.x;
    const int row0 = blockIdx.x * 32;

    // stage W (64 x K) into LDS, coalesced
    for (int i = tid; i < 64 * K; i += 256) {
        sB[(i / K) * LDK + (i % K)] = W[i];
    }
    // stage A tile (32 x K)
    for (int i = tid; i < 32 * K; i += 256) {
        int r = i / K, c = i % K;
        sA[r * LDK + c] = X[(size_t)(row0 + r) * K + c];
    }
    __syncthreads();

    const int wave = tid >> 5;
    const int lane = tid & 31;
    const int mt = (wave >> 2) * 16;       // 0 | 16
    const int nt = (wave & 3) * 16;        // 0,16,32,48
    const int rn = lane & 15;              // m for A, n for B
    const int kp = (lane >> 4) * 2;        // K-pair select

    const float* aRow = &sA[(mt + rn) * LDK + kp];
    const float* bRow = &sB[(nt + rn) * LDK + kp];

    v8f acc = {};
#pragma unroll
    for (int k = 0; k < K; k += 4) {
        float2 af = *(const float2*)(aRow + k);
        float2 bf = *(const float2*)(bRow + k);
        v2f a; a[0] = af.x; a[1] = af.y;
        v2f b; b[0] = bf.x; b[1] = bf.y;
        acc = __builtin_amdgcn_wmma_f32_16x16x4_f32(
            /*neg_a=*/false, a, /*neg_b=*/false, b,
            /*c_mod=*/(short)0, acc, /*reuse_a=*/false, /*reuse_b=*/false);
    }

    const int n = nt + rn;
    const int mb = row0 + mt + ((lane >> 4) ? 8 : 0);
#pragma unroll
    for (int r = 0; r < 8; ++r) {
        XW[(size_t)(mb + r) * 64 + n] = acc[r];
    }
}

// ---------------------------------------------------------------------------
// aggr[dst] += (expbuf[e]/s[dst]) * xw[src]   — 32 lanes per edge, float2/lane
// ---------------------------------------------------------------------------
__global__ void edge_scatter(const long long* __restrict__ ei,
                             const float* __restrict__ expbuf,
                             const float* __restrict__ s,
                             const float* __restrict__ xw,
                             float* __restrict__ aggr, int nE) {
    int e = blockIdx.x * 8 + (threadIdx.x >> 5);
    if (e >= nE) return;
    int lane = threadIdx.x & 31;
    int sn = (int)ei[e];
    int dn = (int)ei[(size_t)nE + e];
    float att = expbuf[e] / s[dn];
    float2 v = *(const float2*)(xw + (size_t)sn * 64 + lane * 2);
    float* dp = aggr + (size_t)dn * 64 + lane * 2;
    atomicAdd(dp + 0, att * v.x);
    atomicAdd(dp + 1, att * v.y);
}

// ---------------------------------------------------------------------------
// out = sigmoid(aggr + xw + b)
// ---------------------------------------------------------------------------
__global__ void node_update(const float* __restrict__ aggr,
                            const float* __restrict__ xw,
                            const float* __restrict__ b,
                            float* __restrict__ out, int n) {
    int i = blockIdx.x * blockDim.x + threadIdx.x;
    if (i >= n) return;
    float v = aggr[i] + xw[i] + b[i & 63];
    out[i] = 1.0f / (1.0f + __expf(-v));
}

extern "C" void kernel_launch(void* const* d_in, const int* in_sizes, int n_in,
                              void* d_out, int out_size, void* d_ws, size_t ws_size,
                              hipStream_t stream) {
    const float*     x    = (const float*)d_in[0];      // [100000,128]
    const long long* ei   = (const long long*)d_in[1];  // [2,1600000] int64
    const float*     attr = (const float*)d_in[2];      // [1600000]
    const float*     W1   = (const float*)d_in[3];      // [64,128]
    const float*     b1   = (const float*)d_in[4];      // [64]
    const float*     W2   = (const float*)d_in[5];      // [64,64]
    const float*     b2   = (const float*)d_in[6];      // [64]
    float* out = (float*)d_out;                         // [100000,64]

    float* ws = (float*)d_ws;
    float* s      = ws;                    // 100000
    float* expbuf = ws + 100000;           // 1600000
    float* xw     = ws + 1700000;          // 6400000
    float* aggr   = ws + 8100000;          // 6400000
    float* h1     = ws + 14500000;         // 6400000

    const int NF = N_NODES * 64;

    // zero accumulators (graph-capture-safe)
    hipMemsetAsync(s, 0, (size_t)N_NODES * sizeof(float), stream);
    hipMemsetAsync(aggr, 0, (size_t)NF * sizeof(float), stream);

    // shared attention denominator (identical for both layers)
    edge_expsum<<<(N_EDGES + 255) / 256, 256, 0, stream>>>(attr, ei, expbuf, s, N_EDGES);

    // ---- layer 1 ----
    gemm_xwT<128><<<N_NODES / 32, 256, 0, stream>>>(x, W1, xw, N_NODES);
    edge_scatter<<<(N_EDGES + 7) / 8, 256, 0, stream>>>(ei, expbuf, s, xw, aggr, N_EDGES);
    node_update<<<(NF + 255) / 256, 256, 0, stream>>>(aggr, xw, b1, h1, NF);

    // ---- layer 2 ----
    hipMemsetAsync(aggr, 0, (size_t)NF * sizeof(float), stream);
    gemm_xwT<64><<<N_NODES / 32, 256, 0, stream>>>(h1, W2, xw, N_NODES);
    edge_scatter<<<(N_EDGES + 7) / 8, 256, 0, stream>>>(ei, expbuf, s, xw, aggr, N_EDGES);
    node_update<<<(NF + 255) / 256, 256, 0, stream>>>(aggr, xw, b2, out, NF);
}